// GraphConv_Concat_79388175499439
// MI455X (gfx1250) — compile-verified
//
#include <hip/hip_runtime.h>

// GraphConv with concat, MI455X (gfx1250, wave32).
// Pipeline: zero ws -> degree atomics -> edge aggregation (scaled gather +
// f32 scatter atomics, L2-resident) -> WMMA f32 16x16x4 GEMM with fused
// rsqrt(deg_in) + bias epilogue.
//
// Round 3: fix async-LDS builtin arg types (v4i AS1* / v4i AS3*, per hipcc
// diagnostic), keep LDS-staged row norms + branch-free epilogue.

typedef __attribute__((ext_vector_type(2))) float v2f;
typedef __attribute__((ext_vector_type(8))) float v8f;

#define DFEAT 128      // in_feats == out_feats
#define KDIM  256      // concat dim = 2*DFEAT
#define XSTRIDE 260    // padded LDS row stride (floats): bank = (4r+k)%64, conflict-free

#if defined(__has_builtin)
#if __has_builtin(__builtin_amdgcn_global_load_async_to_lds_b128)
#define HAVE_ASYNC_LDS 1
#endif
#if __has_builtin(__builtin_amdgcn_s_wait_asynccnt)
#define HAVE_WAIT_ASYNC_BUILTIN 1
#endif
#endif

#ifdef __HIP_DEVICE_COMPILE__
__device__ __forceinline__ void wait_asynccnt0() {
#if defined(HAVE_WAIT_ASYNC_BUILTIN)
    __builtin_amdgcn_s_wait_asynccnt(0);
#else
    asm volatile("s_wait_asynccnt 0x0" ::: "memory");
#endif
}
#endif

// ---------------------------------------------------------------- zero ws
__global__ void zero_kernel(float* __restrict__ p, int n) {
    int i = blockIdx.x * blockDim.x + threadIdx.x;
    if (i < n) p[i] = 0.0f;
}

// ---------------------------------------------------------------- degrees
__global__ void degree_kernel(const int* __restrict__ src,
                              const int* __restrict__ dst,
                              float* __restrict__ degO,
                              float* __restrict__ degI, int E) {
    int e = blockIdx.x * blockDim.x + threadIdx.x;
    if (e < E) {
        atomicAdd(degO + src[e], 1.0f);
        atomicAdd(degI + dst[e], 1.0f);
    }
}

// ------------------------------------------------------------ aggregation
// One wave32 per edge. Lane l handles 4 consecutive feats -> one float4
// gather (512B coalesced per wave) + 4 global_atomic_add_f32 (no return).
__global__ void aggregate_kernel(const float* __restrict__ feat,
                                 const float* __restrict__ affine,
                                 const int* __restrict__ src,
                                 const int* __restrict__ dst,
                                 const float* __restrict__ degO,
                                 float* __restrict__ agg, int E) {
    int gt   = blockIdx.x * blockDim.x + threadIdx.x;
    int e    = gt >> 5;         // wave-uniform
    int lane = gt & 31;
    if (e >= E) return;
    int s = src[e];
    int d = dst[e];
    float scale = affine[e] * rsqrtf(fmaxf(degO[s], 1.0f));
    float4 v = ((const float4*)(feat + (long)s * DFEAT))[lane];
    float* ap = agg + (long)d * DFEAT + lane * 4;
    atomicAdd(ap + 0, v.x * scale);
    atomicAdd(ap + 1, v.y * scale);
    atomicAdd(ap + 2, v.z * scale);
    atomicAdd(ap + 3, v.w * scale);
}

// ------------------------------------------------------------------ GEMM
// Block = 256 threads = 8 waves, computes rows [m0, m0+16) x all 128 cols.
// Stage 16x256 concat tile (+ 16 row-norms) into LDS, then each wave does a
// 16x16 tile via 64 x V_WMMA_F32_16X16X4_F32 over K=256.
__global__ void gemm_norm_bias_kernel(const float* __restrict__ feat,
                                      const float* __restrict__ agg,
                                      const float* __restrict__ weight, // [256][128]
                                      const float* __restrict__ bias,   // [128]
                                      const float* __restrict__ degI,
                                      float* __restrict__ out, int N) {
    __shared__ float ldsX[16 * XSTRIDE];   // 16.6 KB, padded stride
    __shared__ float ldsDn[16];            // rsqrt(max(deg_in,1)) per block row

    const int m0  = blockIdx.x * 16;
    const int tid = threadIdx.x;

    // ---- stage A tile: thread -> (row r = tid>>4, cols [(tid&15)*16, +16))
    {
        int r     = tid >> 4;
        int cbase = (tid & 15) * 16;
        int row   = m0 + r;
        if (row >= N) row = N - 1;                       // clamp (stores guarded)
        const float* sp = (cbase < DFEAT)
            ? (feat + (long)row * DFEAT + cbase)
            : (agg  + (long)row * DFEAT + (cbase - DFEAT));
        float* dp = ldsX + r * XSTRIDE + cbase;
#if defined(HAVE_ASYNC_LDS)
        // builtin signature (per hipcc diagnostic): param0 = v4i AS1*, so
        // (global src, lds dst, imm offset, cpol)
        typedef int v4i __attribute__((vector_size(16)));
        typedef __attribute__((address_space(1))) v4i v4i_g;
        typedef __attribute__((address_space(3))) v4i v4i_l;
#pragma unroll
        for (int j = 0; j < 4; ++j)
            __builtin_amdgcn_global_load_async_to_lds_b128(
                (v4i_g*)(sp + 4 * j), (v4i_l*)(dp + 4 * j), 0, 0);
#else
#pragma unroll
        for (int j = 0; j < 4; ++j)
            ((float4*)dp)[j] = ((const float4*)sp)[j];
#endif
        if (tid < 16) {                                  // per-row output norm
            int rr = m0 + tid;
            ldsDn[tid] = (rr < N) ? rsqrtf(fmaxf(degI[rr], 1.0f)) : 0.0f;
        }
#if defined(HAVE_ASYNC_LDS)
        wait_asynccnt0();
#endif
    }
    __syncthreads();

    // ---- per-wave 16x16 tile
    const int wave    = tid >> 5;
    const int lane    = tid & 31;
    const int n0      = wave * 16;
    const int halfsel = lane >> 4;        // 0: lanes 0-15, 1: lanes 16-31
    const int lm      = lane & 15;
    const int koff    = 2 * halfsel;      // A/B K sub-offset for this half-wave

    const float* aRow = ldsX + lm * XSTRIDE + koff;          // 8B-aligned
    const float* Wp   = weight + koff * DFEAT + n0 + lm;

    v8f acc = {};
#pragma unroll 8
    for (int k = 0; k < KDIM; k += 4) {
        v2f a = *(const v2f*)(aRow + k);                     // A[lm][k+koff .. +1]
        v2f b = { Wp[k * DFEAT], Wp[(k + 1) * DFEAT] };      // B[k+koff .. +1][n0+lm]
        // (neg_a, A, neg_b, B, c_mod, C, reuse_a, reuse_b)
        acc = __builtin_amdgcn_wmma_f32_16x16x4_f32(
            false, a, false, b, (short)0, acc, false, false);
    }

    // ---- epilogue: D vgpr j -> row m0 + j + 8*halfsel, col n0 + lm
    const float bval = bias[n0 + lm];
    float* op = out + (long)(m0 + 8 * halfsel) * DFEAT + n0 + lm;
    if (m0 + 16 <= N) {
        // full tile: branch-free stores
#pragma unroll
        for (int j = 0; j < 8; ++j)
            op[(long)j * DFEAT] = acc[j] * ldsDn[j + 8 * halfsel] + bval;
    } else {
        // tail tile (only possible when N % 16 != 0)
#pragma unroll
        for (int j = 0; j < 8; ++j) {
            int row = m0 + j + 8 * halfsel;
            if (row < N)
                op[(long)j * DFEAT] = acc[j] * ldsDn[j + 8 * halfsel] + bval;
        }
    }
}

// ---------------------------------------------------------------- launch
extern "C" void kernel_launch(void* const* d_in, const int* in_sizes, int n_in,
                              void* d_out, int out_size, void* d_ws, size_t ws_size,
                              hipStream_t stream) {
    const float* feat   = (const float*)d_in[0];
    const float* affine = (const float*)d_in[1];
    const float* weight = (const float*)d_in[2];
    const float* bias   = (const float*)d_in[3];
    const int*   src    = (const int*)d_in[4];
    const int*   dst    = (const int*)d_in[5];
    float*       out    = (float*)d_out;

    const int N = in_sizes[0] / DFEAT;   // 100000
    const int E = in_sizes[1];           // 1600000

    // workspace layout: degO[N] | degI[N] | agg[N*128]  (= 52 MB)
    float* degO = (float*)d_ws;
    float* degI = degO + N;
    float* agg  = degI + N;

    const int nz = N * (2 + DFEAT);
    zero_kernel<<<(nz + 255) / 256, 256, 0, stream>>>(degO, nz);

    degree_kernel<<<(E + 255) / 256, 256, 0, stream>>>(src, dst, degO, degI, E);

    // one wave per edge -> 8 edges per 256-thread block
    aggregate_kernel<<<(E + 7) / 8, 256, 0, stream>>>(feat, affine, src, dst,
                                                      degO, agg, E);

    gemm_norm_bias_kernel<<<(N + 15) / 16, 256, 0, stream>>>(feat, agg, weight,
                                                             bias, degI, out, N);
}